// RPN_6622839571221
// MI455X (gfx1250) — compile-verified
//
#include <hip/hip_runtime.h>
#include <math.h>

// MI455X / gfx1250 (CDNA5, wave32). Dominant work: 70 GFLOP fp32 3x3 conv ->
// implicit GEMM on V_WMMA_F32_16X16X4_F32 (exact fp32). Everything else is
// bandwidth-trivial (<70 MB total at 23.3 TB/s).

typedef float v2f __attribute__((ext_vector_type(2)));
typedef float v8f __attribute__((ext_vector_type(8)));

#define HTOT 37
#define WTOT 50
#define BATCH 8
#define CIN 512
#define COUT 512
#define KDIM 4608            // 512*9
#define HW 1850              // 37*50
#define NTOT 14800           // 8*1850
#define GNUM 20
#define ANUM 16650           // 50*37*9

// ---------------------------------------------------------------------------
// Kernel 1: conv3x3 + bias + relu as implicit GEMM with f32 WMMA.
// Block = 256 threads (8 waves). Tile: M=128 (8 waves x 16 rows), N=64
// (4 subtiles of 16 cols). K staged through LDS in chunks of 32 (im2col).
//
// Per-lane WMMA operand layout (ISA 7.12.2, 32-bit A 16x4):
//   A: lane L holds (M=L%16, K=koff..koff+1), koff = (L<16 ? 0 : 2)  -> float2
//   B assumed symmetric: lane L holds (N=L%16, K=koff..koff+1)       -> float2
//   C/D: VGPR r -> M = r + (L<16 ? 0 : 8), N = L%16
// ---------------------------------------------------------------------------
#define KB 32
#define LDS_STRIDE (KB + 2)   // pad 2 floats to break bank conflicts

__global__ __launch_bounds__(256)
void conv3x3_wmma(const float* __restrict__ feats,
                  const float* __restrict__ w3,
                  const float* __restrict__ bias3,
                  float* __restrict__ conv1)
{
    __shared__ float Bs[64 * LDS_STRIDE];

    const int tid  = threadIdx.x;
    const int lane = tid & 31;
    const int wave = tid >> 5;
    const int nl   = lane & 15;
    const int koff = (lane < 16) ? 0 : 2;
    const int moff = (lane < 16) ? 0 : 8;

    const int nbase = blockIdx.x * 64;
    const int mbase = blockIdx.y * 128 + wave * 16;
    const int mlane = mbase + nl;

    v8f acc0 = {}, acc1 = {}, acc2 = {}, acc3 = {};

    // im2col staging coords: each thread owns one n and 8 consecutive kk
    const int st_n  = tid >> 2;          // 0..63
    const int st_kk = (tid & 3) * 8;     // 0,8,16,24
    const int n_g   = nbase + st_n;
    const bool nvalid = (n_g < NTOT);
    int sb = 0, sh = 0, sw = 0;
    if (nvalid) { sb = n_g / HW; int hw = n_g - sb * HW; sh = hw / WTOT; sw = hw - sh * WTOT; }

    for (int kc = 0; kc < KDIM; kc += KB) {
        __syncthreads();
        // ---- stage im2col B chunk: Bs[n_local][kk], K = ci*9 + kh*3 + kw
        #pragma unroll
        for (int j = 0; j < 8; ++j) {
            const int k = kc + st_kk + j;
            float v = 0.0f;
            if (nvalid) {
                const int ci = k / 9;
                const int r  = k - ci * 9;
                const int kh = r / 3;
                const int kw = r - kh * 3;
                const int ih = sh + kh - 1;
                const int iw = sw + kw - 1;
                if (ih >= 0 && ih < HTOT && iw >= 0 && iw < WTOT)
                    v = feats[((size_t)(sb * CIN + ci) * HTOT + ih) * WTOT + iw];
            }
            Bs[st_n * LDS_STRIDE + st_kk + j] = v;
        }
        __syncthreads();

        // ---- 8 k-steps x 4 n-subtiles = 32 WMMA per chunk per wave
        const float* arow = w3 + (size_t)mlane * KDIM + kc + koff;
        #pragma unroll
        for (int kk = 0; kk < KB; kk += 4) {
            v2f a   = *(const v2f*)(arow + kk);
            v2f bt0 = *(const v2f*)(&Bs[(0 * 16 + nl) * LDS_STRIDE + kk + koff]);
            v2f bt1 = *(const v2f*)(&Bs[(1 * 16 + nl) * LDS_STRIDE + kk + koff]);
            v2f bt2 = *(const v2f*)(&Bs[(2 * 16 + nl) * LDS_STRIDE + kk + koff]);
            v2f bt3 = *(const v2f*)(&Bs[(3 * 16 + nl) * LDS_STRIDE + kk + koff]);
            acc0 = __builtin_amdgcn_wmma_f32_16x16x4_f32(false, a, false, bt0, (short)0, acc0, false, false);
            acc1 = __builtin_amdgcn_wmma_f32_16x16x4_f32(false, a, false, bt1, (short)0, acc1, false, false);
            acc2 = __builtin_amdgcn_wmma_f32_16x16x4_f32(false, a, false, bt2, (short)0, acc2, false, false);
            acc3 = __builtin_amdgcn_wmma_f32_16x16x4_f32(false, a, false, bt3, (short)0, acc3, false, false);
        }
    }

    // ---- bias + relu, store to conv1[co][n]
    #pragma unroll
    for (int r = 0; r < 8; ++r) {
        const int co = mbase + r + moff;
        const float bi = bias3[co];
        const size_t row = (size_t)co * NTOT;
        const int n0 = nbase + nl;
        if (n0      < NTOT) conv1[row + n0]      = fmaxf(acc0[r] + bi, 0.0f);
        if (n0 + 16 < NTOT) conv1[row + n0 + 16] = fmaxf(acc1[r] + bi, 0.0f);
        if (n0 + 32 < NTOT) conv1[row + n0 + 32] = fmaxf(acc2[r] + bi, 0.0f);
        if (n0 + 48 < NTOT) conv1[row + n0 + 48] = fmaxf(acc3[r] + bi, 0.0f);
    }
}

// ---------------------------------------------------------------------------
// Kernel 2: fused 1x1 convs (18 cls + 36 bbox) + paired softmax, one pass
// over conv1 (30 MB read once). Softmax pairs channels (c, c+9): the
// (B,2,333,50) reshape groups channel c<9 into slot 0 and c+9 into slot 1
// at the same (h,w).
// ---------------------------------------------------------------------------
#define CCHUNK 128

__global__ __launch_bounds__(256)
void rpn_head(const float* __restrict__ conv1,
              const float* __restrict__ wc, const float* __restrict__ bc,
              const float* __restrict__ wb, const float* __restrict__ bb,
              float* __restrict__ out)
{
    __shared__ float lw[54 * CCHUNK];
    const int n = blockIdx.x * 256 + threadIdx.x;

    float acc[54];
    #pragma unroll
    for (int c = 0; c < 54; ++c) acc[c] = 0.0f;

    for (int cc = 0; cc < CIN; cc += CCHUNK) {
        __syncthreads();
        for (int i = threadIdx.x; i < 54 * CCHUNK; i += 256) {
            const int c = i / CCHUNK, cl = i - c * CCHUNK;
            lw[i] = (c < 18) ? wc[c * CIN + cc + cl] : wb[(c - 18) * CIN + cc + cl];
        }
        __syncthreads();
        if (n < NTOT) {
            #pragma unroll 4
            for (int cl = 0; cl < CCHUNK; ++cl) {
                const float x = conv1[(size_t)(cc + cl) * NTOT + n];
                #pragma unroll
                for (int c = 0; c < 54; ++c)
                    acc[c] = fmaf(x, lw[c * CCHUNK + cl], acc[c]);
            }
        }
    }
    if (n >= NTOT) return;

    const int b = n / HW, hw = n - b * HW;
    float* ob = out + (size_t)b * 54 * HW + hw;

    #pragma unroll
    for (int j = 0; j < 9; ++j) {
        const float s0 = acc[j] + bc[j];
        const float s1 = acc[j + 9] + bc[j + 9];
        const float m  = fmaxf(s0, s1);
        const float e0 = __expf(s0 - m), e1 = __expf(s1 - m);
        const float inv = 1.0f / (e0 + e1);
        ob[(size_t)j * HW]       = e0 * inv;
        ob[(size_t)(j + 9) * HW] = e1 * inv;
    }
    #pragma unroll
    for (int j = 0; j < 36; ++j)
        ob[(size_t)(18 + j) * HW] = acc[18 + j] + bb[j];
}

// ---------------------------------------------------------------------------
// Anchors / IoU / labels
// anchor index a = ((x*37 + y)*3 + scale)*3 + type, scales {8,16,32}
// ---------------------------------------------------------------------------
__device__ inline void anchor_box(int a, const float* __restrict__ im_info,
                                  float& x1, float& y1, float& x2, float& y2,
                                  bool& inside)
{
    const int ti = a % 3;
    int t = a / 3;
    const int si = t % 3;
    t /= 3;
    const int y = t % HTOT;
    const int x = t / HTOT;
    const float fs = (float)(8 << si);
    const float fx = (float)x, fy = (float)y;
    float hx, hy;
    if (ti == 0)      { hx = fs * 0.50f; hy = fs * 0.50f; }
    else if (ti == 1) { hx = fs;         hy = fs * 0.25f; }
    else              { hx = fs * 0.25f; hy = fs;         }
    x1 = (fx - hx) * 16.0f;
    y1 = (fy - hy) * 16.0f;
    x2 = (fx + hx) * 16.0f;
    y2 = (fy + hy) * 16.0f;
    inside = (x1 >= 0.0f) && (y1 >= 0.0f) && (x2 < im_info[1]) && (y2 < im_info[0]);
}

__device__ inline float iou_one(float ax1, float ay1, float ax2, float ay2,
                                float area_a, const float* __restrict__ g, int j)
{
    const float gx1 = g[j * 4 + 0], gy1 = g[j * 4 + 1];
    const float gx2 = g[j * 4 + 2], gy2 = g[j * 4 + 3];
    const float area_g = (gx2 - gx1) * (gy2 - gy1);
    const float l  = fmaxf(ax1, gx1), r  = fminf(ax2, gx2);
    const float tp = fmaxf(ay1, gy1), bo = fminf(ay2, gy2);
    if (l >= r || tp >= bo) return 0.0f;
    const float inter = (r - l) * (bo - tp);
    return inter / (area_a + area_g - inter);
}

__global__ __launch_bounds__(256)
void iou_kernel(const float* __restrict__ gt, const float* __restrict__ im_info,
                float* __restrict__ iou_out, float* __restrict__ labels)
{
    const int idx = blockIdx.x * 256 + threadIdx.x;
    if (idx >= BATCH * ANUM) return;
    const int b = idx / ANUM, a = idx - b * ANUM;

    float ax1, ay1, ax2, ay2; bool inside;
    anchor_box(a, im_info, ax1, ay1, ax2, ay2, inside);
    const float area_a = (ax2 - ax1) * (ay2 - ay1);

    float* orow = iou_out + (size_t)idx * GNUM;
    const float* g = gt + (size_t)b * GNUM * 4;
    #pragma unroll 4
    for (int j = 0; j < GNUM; ++j) {
        const float v = iou_one(ax1, ay1, ax2, ay2, area_a, g, j);
        orow[j] = inside ? v : 0.0f;
    }
    labels[idx] = 0.0f;
}

__global__ void label_kernel(const float* __restrict__ gt,
                             const float* __restrict__ im_info,
                             float* __restrict__ labels)
{
    const int b = threadIdx.x;
    if (b >= BATCH) return;
    const float* g = gt + (size_t)b * GNUM * 4;

    // argmax over gt of masked iou[b, anchor=GNUM-1, :] (first max, like jnp)
    float best = -1.0f; int bidx = 0;
    {
        float ax1, ay1, ax2, ay2; bool ins;
        anchor_box(GNUM - 1, im_info, ax1, ay1, ax2, ay2, ins);
        const float area_a = (ax2 - ax1) * (ay2 - ay1);
        for (int j = 0; j < GNUM; ++j) {
            float v = iou_one(ax1, ay1, ax2, ay2, area_a, g, j);
            v = ins ? v : 0.0f;
            if (v > best) { best = v; bidx = j; }
        }
    }
    // suppression test at anchor index bidx: inside & max_iou > 0.7 -> 0
    float ax1, ay1, ax2, ay2; bool ins;
    anchor_box(bidx, im_info, ax1, ay1, ax2, ay2, ins);
    const float area_a = (ax2 - ax1) * (ay2 - ay1);
    float mx = 0.0f;
    for (int j = 0; j < GNUM; ++j) {
        float v = iou_one(ax1, ay1, ax2, ay2, area_a, g, j);
        v = ins ? v : 0.0f;
        mx = fmaxf(mx, v);
    }
    const bool supp = ins && (mx > 0.7f);
    labels[(size_t)b * ANUM + bidx] = supp ? 0.0f : 1.0f;
}

// ---------------------------------------------------------------------------
extern "C" void kernel_launch(void* const* d_in, const int* in_sizes, int n_in,
                              void* d_out, int out_size, void* d_ws, size_t ws_size,
                              hipStream_t stream)
{
    const float* feats  = (const float*)d_in[0];
    const float* w3     = (const float*)d_in[1];
    const float* b3     = (const float*)d_in[2];
    const float* wc     = (const float*)d_in[3];
    const float* bc     = (const float*)d_in[4];
    const float* wb     = (const float*)d_in[5];
    const float* bb     = (const float*)d_in[6];
    const float* gt     = (const float*)d_in[7];
    const float* iminfo = (const float*)d_in[8];

    float* out = (float*)d_out;
    float* out_head = out;                              // 8*54*1850 = 799200
    float* iou_out  = out + 799200;                     // 8*16650*20 = 2664000
    float* labels   = out + 799200 + 2664000;           // 8*16650   = 133200

    float* conv1 = (float*)d_ws;                        // 512*14800 fp32 = 30.3 MB

    dim3 gA((NTOT + 63) / 64, COUT / 128);              // (232, 4)
    conv3x3_wmma<<<gA, dim3(256), 0, stream>>>(feats, w3, b3, conv1);

    rpn_head<<<dim3((NTOT + 255) / 256), dim3(256), 0, stream>>>(
        conv1, wc, bc, wb, bb, out_head);

    iou_kernel<<<dim3((BATCH * ANUM + 255) / 256), dim3(256), 0, stream>>>(
        gt, iminfo, iou_out, labels);

    label_kernel<<<dim3(1), dim3(32), 0, stream>>>(gt, iminfo, labels);
}